// Net_8650064134725
// MI455X (gfx1250) — compile-verified
//
#include <hip/hip_runtime.h>

// ---------------------------------------------------------------------------
// MI455X (gfx1250): conv stack + MoE, fp32 WMMA (16x16x4) everywhere.
// Compute-bound (~90 GFLOP; HBM traffic ~1 GB => ~45us at 23.3 TB/s), so all
// GEMM-shaped ops run on v_wmma_f32_16x16x4_f32 with LDS-staged panels.
// Conv: rs-major K ordering hoists im2col addressing out of the WMMA loop.
// GEMM: A panel staged via double-buffered global_load_async_to_lds_b128
// (ASYNCcnt), overlapping the next panel fetch with the current WMMA loop.
// ---------------------------------------------------------------------------

typedef float v2f __attribute__((ext_vector_type(2)));
typedef float v8f __attribute__((ext_vector_type(8)));

#define BN_EPS 1e-5f
static constexpr int BATCHN = 1024;

__device__ __forceinline__ v8f wmma4(v2f a, v2f b, v8f c) {
  // D = A(16x4, f32) * B(4x16, f32) + C(16x16, f32)
  return __builtin_amdgcn_wmma_f32_16x16x4_f32(
      /*neg_a=*/false, a, /*neg_b=*/false, b,
      /*c_mod=*/(short)0, c, /*reuse_a=*/false, /*reuse_b=*/false);
}

__device__ __forceinline__ int imin(int a, int b) { return a < b ? a : b; }
__device__ __forceinline__ int imax(int a, int b) { return a > b ? a : b; }

// async global->LDS copy, 16B per lane, tracked by ASYNCcnt
// (generic LDS pointer: low 32 bits are the LDS byte offset)
__device__ __forceinline__ void async_ld_b128(const float* gptr, float* lptr) {
  const unsigned loff = (unsigned)(unsigned long long)(void*)lptr;
  asm volatile("global_load_async_to_lds_b128 %0, %1, off"
               :: "v"(loff), "v"(gptr)
               : "memory");
}
__device__ __forceinline__ void wait_async0() {
  asm volatile("s_wait_asynccnt 0x0" ::: "memory");
}

// ---------------------------------------------------------------------------
// Implicit-GEMM conv 3x3 SAME + bias + BN(eval) + ReLU, NCHW fp32.
//   M = BATCH*HW*HW pixels, N = COUT, K reordered as rs*CINP + cin.
// 8 waves/block over M; each wave owns NTW 16-wide N-tiles (one A gather
// feeds NTW WMMAs). Weights for one rs position staged in LDS (permuted);
// per-rs halo check hoisted out of the cin loop; halo zeros via clamped
// in-bounds load + select. Pad slots (CIN=3) zero-filled once, race-free.
// ---------------------------------------------------------------------------
template <int CIN, int COUT, int HW, int NTW>
__global__ __launch_bounds__(256) void conv_bn_relu_k(
    const float* __restrict__ x, const float* __restrict__ wgt,
    const float* __restrict__ bias, const float* __restrict__ gamma,
    const float* __restrict__ beta, const float* __restrict__ mean,
    const float* __restrict__ var, float* __restrict__ out) {
  constexpr int HW2 = HW * HW;
  constexpr int K = CIN * 9;
  constexpr int CINP = (CIN + 3) & ~3;   // cin padded to multiple of 4
  constexpr int NW = 16 * NTW;           // output channels per block
  constexpr int LDSTR = NW + 1;          // LDS row stride (pad)
  __shared__ float Bs[CINP * LDSTR];

  const int tid = threadIdx.x;
  const int lane = tid & 31;
  const int wv = tid >> 5;                       // 0..7
  const int m_base = (blockIdx.x * 8 + wv) * 16;
  const int n_base = blockIdx.y * NW;
  const int ln = lane & 15;
  const int kh = lane >> 4;                      // K-half of the fragment

  // A-fragment pixel (row m = m_base + ln)
  const int p = m_base + ln;
  const int bb = p / HW2;
  const int hw = p % HW2;
  const int hh = hw / HW;
  const int ww = hw % HW;
  const float* __restrict__ xb = x + (long long)bb * CIN * HW2;

  // zero-fill pad slots once (disjoint from load pass, no race)
  if constexpr (CINP != CIN) {
    for (int i = tid; i < (CINP - CIN) * NW; i += 256) {
      const int cin = CIN + i % (CINP - CIN);
      const int n = i / (CINP - CIN);
      Bs[cin * LDSTR + n] = 0.0f;
    }
  }

  v8f acc[NTW];
#pragma unroll
  for (int t = 0; t < NTW; ++t) acc[t] = (v8f){0.f,0.f,0.f,0.f,0.f,0.f,0.f,0.f};

  int r = 0, s = 0;
  for (int rs = 0; rs < 9; ++rs) {
    // stage permuted weight panel: Bs[cin][n] = wgt[(n_base+n)*K + cin*9 + rs]
    for (int i = tid; i < CIN * NW; i += 256) {
      const int cin = i % CIN;
      const int n = i / CIN;
      Bs[cin * LDSTR + n] = wgt[(long long)(n_base + n) * K + cin * 9 + rs];
    }
    __syncthreads();

    // per-rs halo handling, hoisted out of the cin loop
    const int ih = hh + r - 1, iw = ww + s - 1;
    const bool valid =
        ((unsigned)ih < (unsigned)HW) && ((unsigned)iw < (unsigned)HW);
    const int ihc = imin(imax(ih, 0), HW - 1);
    const int iwc = imin(imax(iw, 0), HW - 1);
    const float* __restrict__ bp = xb + ihc * HW + iwc;  // cin stride = HW2

#pragma unroll
    for (int c0 = 0; c0 < CINP; c0 += 4) {
      const int ca = c0 + kh * 2;
      // always-in-bounds loads (clamped), zeros selected afterwards
      const int c_a0 = (CINP == CIN) ? ca : imin(ca, CIN - 1);
      const int c_a1 = (CINP == CIN) ? (ca + 1) : imin(ca + 1, CIN - 1);
      const float l0 = bp[(long long)c_a0 * HW2];
      const float l1 = bp[(long long)c_a1 * HW2];
      v2f a;
      a.x = (valid && (CINP == CIN || ca < CIN)) ? l0 : 0.0f;
      a.y = (valid && (CINP == CIN || ca + 1 < CIN)) ? l1 : 0.0f;
#pragma unroll
      for (int t = 0; t < NTW; ++t) {
        v2f b;
        b.x = Bs[ca * LDSTR + 16 * t + ln];
        b.y = Bs[(ca + 1) * LDSTR + 16 * t + ln];
        acc[t] = wmma4(a, b, acc[t]);
      }
    }
    __syncthreads();
    if (++s == 3) { s = 0; ++r; }
  }

  // fused epilogue: y = relu(conv*sc + sh); 8 rows per lane are one
  // contiguous run of 8 w-positions (HW multiple of 8, tile 8-aligned).
  const int m0 = m_base + 8 * kh;
  const int b2 = m0 / HW2;
  const int hw2 = m0 % HW2;
  const int h2 = hw2 / HW;
  const int w0 = hw2 % HW;
#pragma unroll
  for (int t = 0; t < NTW; ++t) {
    const int n = n_base + 16 * t + ln;
    const float sc = gamma[n] * rsqrtf(var[n] + BN_EPS);
    const float sh = beta[n] + (bias[n] - mean[n]) * sc;
    float* __restrict__ op =
        out + ((long long)(b2 * COUT + n) * HW + h2) * HW + w0;
#pragma unroll
    for (int v = 0; v < 8; ++v) {
      const float val = acc[t][v] * sc + sh;
      op[v] = val > 0.0f ? val : 0.0f;
    }
  }
}

// ---------------------------------------------------------------------------
// 2x2 max-pool, NCHW, stride 2.
// ---------------------------------------------------------------------------
template <int C, int H>
__global__ __launch_bounds__(256) void maxpool2_k(const float* __restrict__ in,
                                                  float* __restrict__ out) {
  constexpr int HO = H / 2;
  const long long total = (long long)BATCHN * C * HO * HO;
  const long long idx = (long long)blockIdx.x * blockDim.x + threadIdx.x;
  if (idx >= total) return;
  const int w = idx % HO;
  long long t = idx / HO;
  const int h = t % HO; t /= HO;
  const int c = t % C;
  const int b = t / C;
  const float* base = in + (((long long)(b * C + c) * H + 2 * h) * H + 2 * w);
  const float m0 = fmaxf(base[0], base[1]);
  const float m1 = fmaxf(base[H], base[H + 1]);
  out[idx] = fmaxf(m0, m1);
}

// ---------------------------------------------------------------------------
// fp32 WMMA GEMM: out[M,N] = act( bn( A[M,K] @ Bw[K,N] + bias ) )
//   Bw row-major [K][N] (jax layout). K must be a multiple of 64.
//   A panel (16x64) staged via double-buffered async global->LDS b128 copies
//   (ASYNCcnt): fetch of chunk c+1 overlaps the WMMA loop of chunk c; one
//   barrier per chunk. B loads are lane-coalesced from global/L2.
//   gridDim.z batches experts via strides. NGUARD masks an N tail (<16).
// ---------------------------------------------------------------------------
template <int WAVES, bool DO_BN, bool DO_RELU, bool NGUARD>
__global__ __launch_bounds__(WAVES * 32) void gemm_k(
    const float* __restrict__ A, const float* __restrict__ Bw,
    const float* __restrict__ bias, const float* __restrict__ gamma,
    const float* __restrict__ beta, const float* __restrict__ mean,
    const float* __restrict__ var, float* __restrict__ out,
    int M, int N, int K, int ldOut,
    long long strideA, long long strideB, long long strideP,
    long long strideOut) {
  constexpr int KC = 64;
  constexpr int ASTR = 68;  // 16B-aligned rows (272B), conflict-free reads
  __shared__ __align__(16) float As[2][16 * ASTR];

  const long long e = blockIdx.z;
  A += e * strideA;
  Bw += e * strideB;
  out += e * strideOut;
  bias += e * strideP;
  if constexpr (DO_BN) {
    gamma += e * strideP; beta += e * strideP;
    mean += e * strideP;  var += e * strideP;
  }

  const int tid = threadIdx.x;
  const int lane = tid & 31;
  const int wv = tid >> 5;
  const int m_base = blockIdx.x * 16;
  const int n_base = (blockIdx.y * WAVES + wv) * 16;
  const int ln = lane & 15;
  const int kh = lane >> 4;
  const int n = n_base + ln;
  const bool nvalid = NGUARD ? (n < N) : true;
  const int nc = NGUARD ? imin(n, N - 1) : n;  // clamped (always in-bounds)

  // async-stage one 16xKC A chunk into buffer `buf`
  auto stage = [&](int buf, int kc) {
    for (int i = tid; i < 256; i += WAVES * 32) {
      const int row = i >> 4;
      const int col4 = (i & 15) * 4;
      async_ld_b128(&A[(long long)(m_base + row) * K + kc + col4],
                    &As[buf][row * ASTR + col4]);
    }
  };

  v8f acc = {0.f, 0.f, 0.f, 0.f, 0.f, 0.f, 0.f, 0.f};

  const int nchunks = K / KC;
  stage(0, 0);
  for (int c = 0; c < nchunks; ++c) {
    wait_async0();       // my async fill of chunk c landed in LDS
    __syncthreads();     // everyone's landed; buffer (c+1)&1 is free
    if (c + 1 < nchunks) {
      stage((c + 1) & 1, (c + 1) * KC);
      __builtin_prefetch(&Bw[(long long)(c + 1) * KC * N + nc], 0, 1);
    }
    const float* __restrict__ Asc = &As[c & 1][0];
    const int kc = c * KC;
#pragma unroll
    for (int kk = 0; kk < KC; kk += 4) {
      const int k = kc + kk + kh * 2;
      v2f a, b;
      a.x = Asc[ln * ASTR + kk + kh * 2];
      a.y = Asc[ln * ASTR + kk + kh * 2 + 1];
      const float b0 = Bw[(long long)k * N + nc];
      const float b1 = Bw[(long long)(k + 1) * N + nc];
      if constexpr (NGUARD) {
        b.x = nvalid ? b0 : 0.0f;
        b.y = nvalid ? b1 : 0.0f;
      } else {
        b.x = b0; b.y = b1;
      }
      acc = wmma4(a, b, acc);
    }
  }

  float sc = 1.0f, sh = 0.0f;
  if constexpr (DO_BN) {
    sc = gamma[nc] * rsqrtf(var[nc] + BN_EPS);
    sh = beta[nc] + (bias[nc] - mean[nc]) * sc;
  } else {
    sh = bias[nc];
  }
#pragma unroll
  for (int v = 0; v < 8; ++v) {
    const int m = m_base + v + 8 * kh;
    float val = acc[v] * sc + sh;
    if constexpr (DO_RELU) val = val > 0.0f ? val : 0.0f;
    if (nvalid) out[(long long)m * ldOut + n] = val;
  }
}

// ---------------------------------------------------------------------------
// top-2 over 8 logits + softmax -> dense gates[B,8]
// ---------------------------------------------------------------------------
__global__ __launch_bounds__(256) void topk_gate_k(const float* __restrict__ logits,
                                                   float* __restrict__ gates) {
  const int b = blockIdx.x * blockDim.x + threadIdx.x;
  if (b >= BATCHN) return;
  float v[8];
#pragma unroll
  for (int e = 0; e < 8; ++e) v[e] = logits[b * 8 + e];
  int i0 = 0; float v0 = v[0];
#pragma unroll
  for (int e = 1; e < 8; ++e)
    if (v[e] > v0) { v0 = v[e]; i0 = e; }
  int i1 = -1; float v1 = -3.4e38f;
#pragma unroll
  for (int e = 0; e < 8; ++e)
    if (e != i0 && v[e] > v1) { v1 = v[e]; i1 = e; }
  const float p0 = 1.0f / (1.0f + expf(v1 - v0));
  const float p1 = 1.0f - p0;
#pragma unroll
  for (int e = 0; e < 8; ++e) gates[b * 8 + e] = 0.0f;
  gates[b * 8 + i0] = p0;
  gates[b * 8 + i1] = p1;
}

// out[b,o] = sum_e gates[b,e] * eo[e,b,o]
__global__ __launch_bounds__(256) void combine_k(const float* __restrict__ eo,
                                                 const float* __restrict__ gates,
                                                 float* __restrict__ out) {
  const int idx = blockIdx.x * blockDim.x + threadIdx.x;
  if (idx >= BATCHN * 10) return;
  const int o = idx % 10;
  const int b = idx / 10;
  float s = 0.0f;
#pragma unroll
  for (int e = 0; e < 8; ++e)
    s += gates[b * 8 + e] * eo[((long long)e * BATCHN + b) * 10 + o];
  out[idx] = s;
}

// ---------------------------------------------------------------------------
// launch
// ---------------------------------------------------------------------------
extern "C" void kernel_launch(void* const* d_in, const int* in_sizes, int n_in,
                              void* d_out, int out_size, void* d_ws,
                              size_t ws_size, hipStream_t stream) {
  (void)in_sizes; (void)n_in; (void)out_size; (void)ws_size;

  const float* x = (const float*)d_in[0];
  auto cp = [&](int layer, int j) -> const float* {
    return (const float*)d_in[1 + layer * 6 + j];
  };
  const float* ew1 = (const float*)d_in[37];
  const float* eb1 = (const float*)d_in[38];
  const float* eg1 = (const float*)d_in[39];
  const float* ebe1 = (const float*)d_in[40];
  const float* em1 = (const float*)d_in[41];
  const float* ev1 = (const float*)d_in[42];
  const float* ew2 = (const float*)d_in[43];
  const float* eb2 = (const float*)d_in[44];
  const float* eg2 = (const float*)d_in[45];
  const float* ebe2 = (const float*)d_in[46];
  const float* em2 = (const float*)d_in[47];
  const float* ev2 = (const float*)d_in[48];
  const float* ew3 = (const float*)d_in[49];
  const float* eb3 = (const float*)d_in[50];
  const float* gw1 = (const float*)d_in[51];
  const float* gb1 = (const float*)d_in[52];
  const float* gg = (const float*)d_in[53];
  const float* gbe = (const float*)d_in[54];
  const float* gm = (const float*)d_in[55];
  const float* gv = (const float*)d_in[56];
  const float* gw2 = (const float*)d_in[57];
  const float* gb2 = (const float*)d_in[58];

  float* wsf = (float*)d_ws;
  const long long NA = 33554432LL;  // 1024*32*32*32 floats, ping/pong halves
  float* bufA = wsf;
  float* bufB = wsf + NA;

  // ---- conv stack (implicit-GEMM WMMA, fused bias+BN+ReLU) ----
  conv_bn_relu_k<3, 32, 32, 2><<<dim3(8192, 1), 256, 0, stream>>>(
      x, cp(0, 0), cp(0, 1), cp(0, 2), cp(0, 3), cp(0, 4), cp(0, 5), bufA);
  conv_bn_relu_k<32, 32, 32, 2><<<dim3(8192, 1), 256, 0, stream>>>(
      bufA, cp(1, 0), cp(1, 1), cp(1, 2), cp(1, 3), cp(1, 4), cp(1, 5), bufB);
  maxpool2_k<32, 32><<<32768, 256, 0, stream>>>(bufB, bufA);
  conv_bn_relu_k<32, 64, 16, 4><<<dim3(2048, 1), 256, 0, stream>>>(
      bufA, cp(2, 0), cp(2, 1), cp(2, 2), cp(2, 3), cp(2, 4), cp(2, 5), bufB);
  conv_bn_relu_k<64, 64, 16, 4><<<dim3(2048, 1), 256, 0, stream>>>(
      bufB, cp(3, 0), cp(3, 1), cp(3, 2), cp(3, 3), cp(3, 4), cp(3, 5), bufA);
  maxpool2_k<64, 16><<<16384, 256, 0, stream>>>(bufA, bufB);
  conv_bn_relu_k<64, 128, 8, 4><<<dim3(512, 2), 256, 0, stream>>>(
      bufB, cp(4, 0), cp(4, 1), cp(4, 2), cp(4, 3), cp(4, 4), cp(4, 5), bufA);
  conv_bn_relu_k<128, 128, 8, 4><<<dim3(512, 2), 256, 0, stream>>>(
      bufA, cp(5, 0), cp(5, 1), cp(5, 2), cp(5, 3), cp(5, 4), cp(5, 5), bufB);
  maxpool2_k<128, 8><<<8192, 256, 0, stream>>>(bufB, bufA);

  const float* f = bufA;  // [1024, 2048] (C,H,W flatten matches reference)

  float* gate_h = bufB;                   // [1024,128]
  float* logits = gate_h + 1024LL * 128;  // [1024,8]
  float* gates = logits + 1024LL * 8;     // [1024,8]
  float* h1 = gates + 1024LL * 8;         // [8,1024,256]
  float* h2 = h1 + 8LL * 1024 * 256;      // [8,1024,256]
  float* eo = h2 + 8LL * 1024 * 256;      // [8,1024,10]

  // ---- gate MLP ----
  gemm_k<8, true, true, false><<<dim3(64, 1, 1), 256, 0, stream>>>(
      f, gw1, gb1, gg, gbe, gm, gv, gate_h,
      1024, 128, 2048, 128, 0, 0, 0, 0);
  gemm_k<1, false, false, true><<<dim3(64, 1, 1), 32, 0, stream>>>(
      gate_h, gw2, gb2, gb2, gb2, gb2, gb2, logits,
      1024, 8, 128, 8, 0, 0, 0, 0);
  topk_gate_k<<<4, 256, 0, stream>>>(logits, gates);

  // ---- experts (dense over E=8, batched via gridDim.z) ----
  gemm_k<8, true, true, false><<<dim3(64, 2, 8), 256, 0, stream>>>(
      f, ew1, eb1, eg1, ebe1, em1, ev1, h1,
      1024, 256, 2048, 256, 0, 2048LL * 256, 256, 1024LL * 256);
  gemm_k<8, true, true, false><<<dim3(64, 2, 8), 256, 0, stream>>>(
      h1, ew2, eb2, eg2, ebe2, em2, ev2, h2,
      1024, 256, 256, 256, 1024LL * 256, 256LL * 256, 256, 1024LL * 256);
  gemm_k<1, false, false, true><<<dim3(64, 1, 8), 32, 0, stream>>>(
      h2, ew3, eb3, eb3, eb3, eb3, eb3, eo,
      1024, 10, 256, 10, 1024LL * 256, 256LL * 10, 10, 1024LL * 10);

  // ---- weighted combine ----
  combine_k<<<40, 256, 0, stream>>>(eo, gates, (float*)d_out);
}